// TemporalFusion_86758339379212
// MI455X (gfx1250) — compile-verified
//
#include <hip/hip_runtime.h>
#include <hip/hip_bf16.h>
#include <math.h>

#define T_  20
#define B_  48
#define CH_ 16
#define H_  16
#define W_  264
#define HW_ 4224           // H_*W_
#define D_  67584          // CH_*H_*W_ == L*C
#define NCHUNK 64
#define CHUNK_ (D_ / NCHUNK)   // 1056, multiple of 4

typedef float v2f __attribute__((ext_vector_type(2)));
typedef float v8f __attribute__((ext_vector_type(8)));

// ---------------------------------------------------------------- zero G
__global__ __launch_bounds__(256) void tf_zero_k(float* __restrict__ g, int n) {
    int i = blockIdx.x * 256 + threadIdx.x;
    if (i < n) g[i] = 0.f;
}

// ------------------------------------------------- poses[b][j] = sum_{k>=j} delta[b+1+k]
__global__ __launch_bounds__(256) void tf_pose_k(const float* __restrict__ delta,
                                                 float* __restrict__ poses) {
    int idx = blockIdx.x * 256 + threadIdx.x;
    if (idx >= B_ * (T_ - 1) * 3) return;
    int c  = idx % 3;
    int bj = idx / 3;
    int j  = bj % (T_ - 1);
    int b  = bj / (T_ - 1);
    float acc = 0.f;
    for (int k = j; k < T_ - 1; ++k) acc += delta[(b + 1 + k) * 3 + c];
    poses[idx] = acc;
}

// ------------------------------------------------- build flat[b][t][:] (warp or copy)
__global__ __launch_bounds__(256) void tf_warp_k(const float* __restrict__ bev,
                                                 const float* __restrict__ poses,
                                                 float* __restrict__ flat) {
    const int p = blockIdx.x * 256 + threadIdx.x;
    if (p >= HW_) return;
    const int t = blockIdx.y, b = blockIdx.z;
    float* dst = flat + ((size_t)b * T_ + t) * D_ + p;

    if (t == T_ - 1) {              // aligned[:, T-1] = current frame
        const float* src = bev + (size_t)b * D_ + p;
        #pragma unroll
        for (int c = 0; c < CH_; ++c) dst[c * HW_] = src[c * HW_];
        return;
    }
    const int j = T_ - 2 - t;       // warped[:, ::-1]
    const float* feat = bev + (size_t)(b + 2 + j) * D_;
    const float* ps   = poses + ((size_t)b * (T_ - 1) + j) * 3;
    const float dx = ps[0], dy = ps[1], yaw = ps[2];
    const float cs = cosf(yaw), sn = sinf(yaw);

    const int hh = p / W_, ww = p - hh * W_;
    const float xs = (2.f * ww + 1.f) / (float)W_ - 1.f;
    const float ys = (2.f * hh + 1.f) / (float)H_ - 1.f;
    const float gx =  cs * xs + sn * ys + dx;
    const float gy = -sn * xs + cs * ys + dy;
    const float ix = ((gx + 1.f) * W_ - 1.f) * 0.5f;
    const float iy = ((gy + 1.f) * H_ - 1.f) * 0.5f;
    const float x0f = floorf(ix), y0f = floorf(iy);
    const float wx1 = ix - x0f, wy1 = iy - y0f;
    const int x0 = (int)x0f, y0 = (int)y0f, x1 = x0 + 1, y1 = y0 + 1;

    float w00 = (1.f - wy1) * (1.f - wx1);
    float w01 = (1.f - wy1) * wx1;
    float w10 = wy1 * (1.f - wx1);
    float w11 = wy1 * wx1;
    w00 = (x0 >= 0 && x0 < W_ && y0 >= 0 && y0 < H_) ? w00 : 0.f;
    w01 = (x1 >= 0 && x1 < W_ && y0 >= 0 && y0 < H_) ? w01 : 0.f;
    w10 = (x0 >= 0 && x0 < W_ && y1 >= 0 && y1 < H_) ? w10 : 0.f;
    w11 = (x1 >= 0 && x1 < W_ && y1 >= 0 && y1 < H_) ? w11 : 0.f;

    const int cx0 = min(max(x0, 0), W_ - 1), cx1 = min(max(x1, 0), W_ - 1);
    const int cy0 = min(max(y0, 0), H_ - 1), cy1 = min(max(y1, 0), H_ - 1);
    const int o00 = cy0 * W_ + cx0, o01 = cy0 * W_ + cx1;
    const int o10 = cy1 * W_ + cx0, o11 = cy1 * W_ + cx1;

    #pragma unroll
    for (int c = 0; c < CH_; ++c) {
        const float* fc = feat + c * HW_;
        dst[c * HW_] = w00 * fc[o00] + w01 * fc[o01] + w10 * fc[o10] + w11 * fc[o11];
    }
}

// ------------------------------------------------- G[b] += F F^T over one K chunk (WMMA f32)
// A-fragment layout (16x4 f32): lane = m + 16*h holds F[row, k+2h], F[row, k+2h+1].
// By symmetry of the N-striped B layout, the same float2 serves as the B operand.
__global__ __launch_bounds__(32) void tf_gram_k(const float* __restrict__ flat,
                                                float* __restrict__ G) {
    const int b     = blockIdx.y;
    const int kbase = blockIdx.x * CHUNK_;
    const int lane  = threadIdx.x;        // 0..31, wave32
    const int m     = lane & 15;
    const int h     = lane >> 4;          // K half-select

    const float* F = flat + (size_t)b * T_ * D_;
    const float* row0 = F + (size_t)m * D_;                       // tile0 rows 0..15 (all valid)
    const int    r1   = min(16 + m, T_ - 1);
    const float  v1   = (16 + m) < T_ ? 1.f : 0.f;                // zero-pad rows 20..31
    const float* row1 = F + (size_t)r1 * D_;

    v8f acc00 = {0.f,0.f,0.f,0.f,0.f,0.f,0.f,0.f};
    v8f acc01 = {0.f,0.f,0.f,0.f,0.f,0.f,0.f,0.f};
    v8f acc11 = {0.f,0.f,0.f,0.f,0.f,0.f,0.f,0.f};

    #pragma unroll 4
    for (int k = kbase; k < kbase + CHUNK_; k += 4) {
        const int off = k + 2 * h;
        v2f a0 = *(const v2f*)(row0 + off);
        v2f a1 = *(const v2f*)(row1 + off);
        a1.x *= v1; a1.y *= v1;
        acc00 = __builtin_amdgcn_wmma_f32_16x16x4_f32(false, a0, false, a0,
                                                      (short)0, acc00, false, false);
        acc01 = __builtin_amdgcn_wmma_f32_16x16x4_f32(false, a0, false, a1,
                                                      (short)0, acc01, false, false);
        acc11 = __builtin_amdgcn_wmma_f32_16x16x4_f32(false, a1, false, a1,
                                                      (short)0, acc11, false, false);
    }

    // C/D layout: element i of lane l -> M = i + 8*(l>>4), N = l&15
    float* Gb = G + (size_t)b * T_ * T_;
    #pragma unroll
    for (int i = 0; i < 8; ++i) {
        const int Mr = i + 8 * h;      // 0..15 within tile
        atomicAdd(&Gb[Mr * T_ + m], acc00[i]);                    // G[Mr][m]
        if (16 + m < T_) {
            atomicAdd(&Gb[Mr * T_ + 16 + m], acc01[i]);           // G[Mr][16+m]
            atomicAdd(&Gb[(16 + m) * T_ + Mr], acc01[i]);         // symmetric tile10
        }
        if (16 + Mr < T_ && 16 + m < T_)
            atomicAdd(&Gb[(16 + Mr) * T_ + 16 + m], acc11[i]);    // G[16+Mr][16+m]
    }
}

// ------------------------------------------------- row softmax, then mean over rows
__global__ __launch_bounds__(32) void tf_softmax_k(const float* __restrict__ G,
                                                   float* __restrict__ wm) {
    __shared__ float P[T_][T_];
    const int b = blockIdx.x, t = threadIdx.x;
    if (t < T_) {
        float row[T_]; float mx = -3.0e38f;
        #pragma unroll
        for (int s = 0; s < T_; ++s) { row[s] = G[(size_t)b*T_*T_ + t*T_ + s]; mx = fmaxf(mx, row[s]); }
        float sum = 0.f;
        #pragma unroll
        for (int s = 0; s < T_; ++s) { row[s] = expf(row[s] - mx); sum += row[s]; }
        const float inv = 1.f / sum;
        #pragma unroll
        for (int s = 0; s < T_; ++s) P[t][s] = row[s] * inv;
    }
    __syncthreads();
    if (t < T_) {
        float a = 0.f;
        #pragma unroll
        for (int tt = 0; tt < T_; ++tt) a += P[tt][t];
        wm[b * T_ + t] = a * (1.f / (float)T_);
    }
}

// ------------------------------------------------- out[b][d] = sum_s wm[b][s] * flat[b][s][d]
__global__ __launch_bounds__(256) void tf_out_k(const float* __restrict__ flat,
                                                const float* __restrict__ wm,
                                                float* __restrict__ out) {
    const int b  = blockIdx.y;
    const int p4 = blockIdx.x * 256 + threadIdx.x;     // 0..D_/4-1
    const float4* F = (const float4*)(flat + (size_t)b * T_ * D_);
    float w[T_];
    #pragma unroll
    for (int s = 0; s < T_; ++s) w[s] = wm[b * T_ + s];
    float4 acc = {0.f, 0.f, 0.f, 0.f};
    #pragma unroll 4
    for (int s = 0; s < T_; ++s) {
        float4 v = F[(size_t)s * (D_ / 4) + p4];
        acc.x += w[s] * v.x; acc.y += w[s] * v.y;
        acc.z += w[s] * v.z; acc.w += w[s] * v.w;
    }
    ((float4*)(out + (size_t)b * D_))[p4] = acc;
}

// ----------------------------------------------------------------
extern "C" void kernel_launch(void* const* d_in, const int* in_sizes, int n_in,
                              void* d_out, int out_size, void* d_ws, size_t ws_size,
                              hipStream_t stream) {
    const float* bev   = (const float*)d_in[0];   // (B+T, L, C) f32
    const float* dpose = (const float*)d_in[1];   // (B+T, 3)   f32
    float* out = (float*)d_out;                   // (B, L, C)  f32

    // workspace layout
    float* flat  = (float*)d_ws;                                  // B*T*D floats (~260 MB)
    float* G     = flat + (size_t)B_ * T_ * D_;                   // B*20*20
    float* wmean = G + (size_t)B_ * T_ * T_;                      // B*20
    float* poses = wmean + (size_t)B_ * T_;                       // B*19*3

    const int gN = B_ * T_ * T_;
    tf_zero_k<<<(gN + 255) / 256, 256, 0, stream>>>(G, gN);

    const int pN = B_ * (T_ - 1) * 3;
    tf_pose_k<<<(pN + 255) / 256, 256, 0, stream>>>(dpose, poses);

    tf_warp_k<<<dim3((HW_ + 255) / 256, T_, B_), 256, 0, stream>>>(bev, poses, flat);

    tf_gram_k<<<dim3(NCHUNK, B_), 32, 0, stream>>>(flat, G);

    tf_softmax_k<<<B_, 32, 0, stream>>>(G, wmean);

    tf_out_k<<<dim3(D_ / 4 / 256, B_), 256, 0, stream>>>(flat, wmean, out);
}